// S6Layer_20882130993585
// MI455X (gfx1250) — compile-verified
//
#include <hip/hip_runtime.h>

// ---------------- problem constants (from reference) ----------------
#define BB   2
#define LL   2048
#define DD   1024
#define NN   16
#define KK   DD                 // GEMM reduction dim
#define MM   (BB * LL)          // 4096 projection rows
#define CH   32                 // chunks along L for the parallel scan
#define CL   (LL / CH)          // 64 timesteps per chunk

// ---- projection kernel tiling (TDM-staged A operand) ----
#define KC            32                   // K columns per TDM chunk
#define NCHUNK        (KK / KC)            // 32 chunks
#define LDS_ROW_DW    (KC + 2)             // 34 DWORDs: 2-DWORD pad per row (bank spread)
#define LDS_TILE_F    (16 * LDS_ROW_DW)    // floats per 16-row tile buffer
#define LDS_TILE_BYTES (LDS_TILE_F * 4)    // 2176 B
#define WAVES_PER_BLOCK 8

typedef __attribute__((ext_vector_type(2))) float v2f;
typedef __attribute__((ext_vector_type(8))) float v8f;
typedef unsigned int u32x4 __attribute__((ext_vector_type(4)));
typedef int          i32x4 __attribute__((ext_vector_type(4)));
typedef int          i32x8 __attribute__((ext_vector_type(8)));

// =====================================================================
// TDM: DMA a 16-row x KC-col fp32 tile (row stride 1024 elems) from
// global memory into LDS, with 2-DWORD padding after each 32-DWORD row.
// D# layout per CDNA5 ISA.
// amdgpu-toolchain (clang-23) builtin form: 6 args
//   (u32x4 g0, i32x8 g1, i32x4 g2, i32x4 g3, i32x8 g4, i32 cpol)
// =====================================================================
__device__ __forceinline__ void tdm_load_tile(const float* gsrc, unsigned lds_off)
{
    unsigned long long ga = (unsigned long long)(uintptr_t)gsrc;
    u32x4 g0;
    g0[0] = 1u;                                            // count=1, user mode
    g0[1] = lds_off;                                       // LDS byte address
    g0[2] = (unsigned)ga;                                  // global_addr[31:0]
    g0[3] = ((unsigned)(ga >> 32) & 0x01FFFFFFu)           // global_addr[56:32]
          | (2u << 30);                                    // type=2 (image)
    i32x8 g1;
    g1[0] = (int)((2u << 16)      // data_size = 4 bytes
                | (1u << 20)      // pad_enable
                | (4u << 22)      // pad_interval: 32 DWORDs
                | (1u << 25));    // pad_amount: 2 DWORDs
    g1[1] = (int)(1024u << 16);   // tensor_dim0 = 1024 (bits 79:48, low half)
    g1[2] = (int)(4096u << 16);   // tensor_dim0 hi=0 | tensor_dim1 = 4096 (low16)
    g1[3] = (int)((unsigned)KC << 16); // tensor_dim1 hi=0 | tile_dim0 = KC
    g1[4] = 16;                   // tile_dim1 = 16 rows | tile_dim2 = 0
    g1[5] = 1024;                 // tensor_dim0_stride[31:0] = 1024 elems
    g1[6] = 0;                    // stride0 hi | stride1 lo
    g1[7] = 0;                    // stride1 hi
    i32x4 g2 = {0, 0, 0, 0};      // 2-D tile: groups 2/3 unused
    i32x4 g3 = {0, 0, 0, 0};
    i32x8 g4 = {0, 0, 0, 0, 0, 0, 0, 0};
    __builtin_amdgcn_tensor_load_to_lds(g0, g1, g2, g3, g4, 0);
}

// =====================================================================
// Kernel 1: fused projections via fp32 WMMA (V_WMMA_F32_16X16X4_F32)
//   Bm = X @ W_B^T + b_B     [MM, 16]
//   Cm = X @ W_C^T + b_C     [MM, 16]
//   Dl = softplus(X @ W_D^T + b_D + bias_D)   [MM]
// One wave per 16-row tile. The A operand is staged through LDS with
// double-buffered TENSOR_LOAD_TO_LDS (TENSORcnt pipelined); fragments
// are read conflict-free thanks to the padded row stride.
// =====================================================================
__global__ __launch_bounds__(256) void s6_proj_wmma(
    const float* __restrict__ X,      // [MM, KK]
    const float* __restrict__ Wb,     // [16, KK]
    const float* __restrict__ bB,     // [16]
    const float* __restrict__ Wc,     // [16, KK]
    const float* __restrict__ bC,     // [16]
    const float* __restrict__ Wd,     // [1, KK]
    const float* __restrict__ bD,     // [1]
    const float* __restrict__ biasD,  // [1]
    float* __restrict__ Bm,           // [MM, 16]
    float* __restrict__ Cm,           // [MM, 16]
    float* __restrict__ Dl)           // [MM]
{
    __shared__ float smem[WAVES_PER_BLOCK * 2 * LDS_TILE_F];  // 34,816 B

    const int lane  = threadIdx.x & 31;
    const int wv    = __builtin_amdgcn_readfirstlane((int)(threadIdx.x >> 5));
    const int tile  = blockIdx.x * WAVES_PER_BLOCK + wv;   // 0..255
    const int row16 = lane & 15;       // M (A frag) / N (B frag, C/D col)
    const int khalf = lane >> 4;       // 0: K pair {0,1}, 1: K pair {2,3}

    const float*   smem_wave = smem + wv * 2 * LDS_TILE_F;
    const unsigned lds_wave  = (unsigned)(size_t)(void*)smem
                             + (unsigned)(wv * 2 * LDS_TILE_BYTES);

    const float* xtile = X  + (size_t)tile * 16 * KK;      // 16 consecutive rows
    const float* wbrow = Wb + (size_t)row16 * KK + khalf * 2;
    const float* wcrow = Wc + (size_t)row16 * KK + khalf * 2;
    const float* wdrow = Wd + khalf * 2;

    v8f accB = {};
    v8f accC = {};
    float dsum = 0.0f;

    tdm_load_tile(xtile, lds_wave);                        // prime buffer 0

    for (int ch = 0; ch < NCHUNK; ++ch) {
        if (ch + 1 < NCHUNK) {
            tdm_load_tile(xtile + (ch + 1) * KC,
                          lds_wave + (unsigned)(((ch + 1) & 1) * LDS_TILE_BYTES));
            __builtin_amdgcn_s_wait_tensorcnt((short)1);   // chunk ch landed
        } else {
            __builtin_amdgcn_s_wait_tensorcnt((short)0);
        }

        const float* abase = smem_wave + (ch & 1) * LDS_TILE_F
                           + row16 * LDS_ROW_DW + khalf * 2;
        const int kg = ch * KC;

        #pragma unroll
        for (int ks = 0; ks < KC; ks += 4) {
            v2f a  = *(const v2f*)(abase + ks);            // ds_load_b64
            v2f wb = *(const v2f*)(wbrow + kg + ks);
            v2f wc = *(const v2f*)(wcrow + kg + ks);
            v2f wd = *(const v2f*)(wdrow + kg + ks);
            // D = A(16x4,f32) x B(4x16,f32) + C(16x16,f32)
            accB = __builtin_amdgcn_wmma_f32_16x16x4_f32(
                       false, a, false, wb, (short)0, accB, false, false);
            accC = __builtin_amdgcn_wmma_f32_16x16x4_f32(
                       false, a, false, wc, (short)0, accC, false, false);
            dsum = fmaf(a.x, wd.x, dsum);
            dsum = fmaf(a.y, wd.y, dsum);
        }
    }

    // Delta: lanes (l, l+16) each hold half the K sum for row m = tile*16+l
    dsum += __shfl_xor(dsum, 16, 32);
    if (lane < 16) {
        float pre = dsum + bD[0] + biasD[0];
        Dl[tile * 16 + row16] = (pre > 20.0f) ? pre : log1pf(__expf(pre));
    }

    // C/D layout: VGPR j -> (M = j + 8*khalf, N = lane%16)
    const int ncol = row16;
    #pragma unroll
    for (int j = 0; j < 8; ++j) {
        int mr = tile * 16 + j + 8 * khalf;
        Bm[mr * NN + ncol] = accB[j] + bB[ncol];
        Cm[mr * NN + ncol] = accC[j] + bC[ncol];
    }
}

// =====================================================================
// Kernel 2 (scan phase A): per-chunk affine summaries.
// s <- a*s + x*bb over a chunk composes to s_out = P*s_in + Q.
// One thread per (b, chunk, d): 65536 threads.
// =====================================================================
__global__ __launch_bounds__(256) void s6_chunk_summary(
    const float* __restrict__ seq,   // [BB*LL, DD]
    const float* __restrict__ A,     // [DD, NN]
    const float* __restrict__ Bm,    // [MM, NN]
    const float* __restrict__ Dl,    // [MM]
    float* __restrict__ P,           // [BB, CH, DD, NN]
    float* __restrict__ Q)           // [BB, CH, DD, NN]
{
    int tid = blockIdx.x * blockDim.x + threadIdx.x;
    int d = tid & (DD - 1);
    int c = (tid >> 10) & (CH - 1);
    int b = tid >> 15;

    float Ar[NN], invA[NN], p[NN], q[NN];
    const float4* arow = (const float4*)(A + (size_t)d * NN);
    #pragma unroll
    for (int v = 0; v < 4; ++v) {
        float4 t = arow[v];
        Ar[4*v+0] = t.x; Ar[4*v+1] = t.y; Ar[4*v+2] = t.z; Ar[4*v+3] = t.w;
    }
    #pragma unroll
    for (int n = 0; n < NN; ++n) {
        invA[n] = 1.0f / Ar[n];
        p[n] = 1.0f;
        q[n] = 0.0f;
    }

    const int l0 = c * CL;
    for (int l = l0; l < l0 + CL; ++l) {
        int idx = b * LL + l;
        float delta = Dl[idx];
        float x = seq[(size_t)idx * DD + d];
        __builtin_prefetch(&seq[(size_t)(idx + 1) * DD + d], 0, 0);
        const float* bmrow = Bm + (size_t)idx * NN;
        #pragma unroll
        for (int n = 0; n < NN; ++n) {
            float a  = __expf(Ar[n] * delta);
            float bb = invA[n] * (a - 1.0f) * bmrow[n];
            p[n] *= a;
            q[n] = fmaf(q[n], a, x * bb);
        }
    }

    size_t base = ((size_t)(b * CH + c) * DD + d) * NN;
    #pragma unroll
    for (int n = 0; n < NN; ++n) {
        P[base + n] = p[n];
        Q[base + n] = q[n];
    }
}

// =====================================================================
// Kernel 3 (scan phase B): exclusive scan over the CH=32 chunks.
// One thread per (b, d, n): 32768 threads, 32 serial steps.
// =====================================================================
__global__ __launch_bounds__(256) void s6_chunk_scan(
    const float* __restrict__ P,
    const float* __restrict__ Q,
    float* __restrict__ Spre)
{
    int tid = blockIdx.x * blockDim.x + threadIdx.x;
    int n = tid & (NN - 1);
    int d = (tid >> 4) & (DD - 1);
    int b = tid >> 14;

    float s = 0.0f;
    for (int c = 0; c < CH; ++c) {
        size_t off = ((size_t)(b * CH + c) * DD + d) * NN + n;
        Spre[off] = s;
        s = fmaf(P[off], s, Q[off]);
    }
}

// =====================================================================
// Kernel 4 (scan phase C): replay each chunk from its prefix state,
// emit y[b,l,d] = sum_n s[n]*Cm[b,l,n].
// =====================================================================
__global__ __launch_bounds__(256) void s6_apply(
    const float* __restrict__ seq,
    const float* __restrict__ A,
    const float* __restrict__ Bm,
    const float* __restrict__ Cm,
    const float* __restrict__ Dl,
    const float* __restrict__ Spre,
    float* __restrict__ out)          // [BB*LL, DD]
{
    int tid = blockIdx.x * blockDim.x + threadIdx.x;
    int d = tid & (DD - 1);
    int c = (tid >> 10) & (CH - 1);
    int b = tid >> 15;

    float Ar[NN], invA[NN], s[NN];
    const float4* arow = (const float4*)(A + (size_t)d * NN);
    #pragma unroll
    for (int v = 0; v < 4; ++v) {
        float4 t = arow[v];
        Ar[4*v+0] = t.x; Ar[4*v+1] = t.y; Ar[4*v+2] = t.z; Ar[4*v+3] = t.w;
    }
    size_t base = ((size_t)(b * CH + c) * DD + d) * NN;
    #pragma unroll
    for (int n = 0; n < NN; ++n) {
        invA[n] = 1.0f / Ar[n];
        s[n] = Spre[base + n];
    }

    const int l0 = c * CL;
    for (int l = l0; l < l0 + CL; ++l) {
        int idx = b * LL + l;
        float delta = Dl[idx];
        float x = seq[(size_t)idx * DD + d];
        __builtin_prefetch(&seq[(size_t)(idx + 1) * DD + d], 0, 0);
        const float* bmrow = Bm + (size_t)idx * NN;
        const float* cmrow = Cm + (size_t)idx * NN;
        float y = 0.0f;
        #pragma unroll
        for (int n = 0; n < NN; ++n) {
            float a  = __expf(Ar[n] * delta);
            float bb = invA[n] * (a - 1.0f) * bmrow[n];
            s[n] = fmaf(s[n], a, x * bb);
            y = fmaf(s[n], cmrow[n], y);
        }
        out[(size_t)idx * DD + d] = y;
    }
}

// =====================================================================
// Host launcher
// =====================================================================
extern "C" void kernel_launch(void* const* d_in, const int* in_sizes, int n_in,
                              void* d_out, int out_size, void* d_ws, size_t ws_size,
                              hipStream_t stream)
{
    const float* seq   = (const float*)d_in[0];
    const float* A     = (const float*)d_in[1];
    const float* W_B   = (const float*)d_in[2];
    const float* b_B   = (const float*)d_in[3];
    const float* W_C   = (const float*)d_in[4];
    const float* b_C   = (const float*)d_in[5];
    const float* W_D   = (const float*)d_in[6];
    const float* b_D   = (const float*)d_in[7];
    const float* biasD = (const float*)d_in[8];
    float* out = (float*)d_out;

    // workspace carve-up (floats)
    const size_t pq = (size_t)BB * CH * DD * NN;   // 1,048,576
    float* ws   = (float*)d_ws;
    float* P    = ws;
    float* Q    = P + pq;
    float* Spre = Q + pq;
    float* Bm   = Spre + pq;
    float* Cm   = Bm + (size_t)MM * NN;
    float* Dl   = Cm + (size_t)MM * NN;
    (void)ws_size; (void)in_sizes; (void)n_in; (void)out_size;

    // 1) projections (WMMA f32 + TDM-staged A operand): 256 tiles, 256 waves
    s6_proj_wmma<<<32, 256, 0, stream>>>(seq, W_B, b_B, W_C, b_C,
                                         W_D, b_D, biasD, Bm, Cm, Dl);
    // 2) per-chunk affine summaries: B*CH*D = 65536 threads
    s6_chunk_summary<<<256, 256, 0, stream>>>(seq, A, Bm, Dl, P, Q);
    // 3) scan over chunks: B*D*N = 32768 threads
    s6_chunk_scan<<<128, 256, 0, stream>>>(P, Q, Spre);
    // 4) replay + output: 65536 threads
    s6_apply<<<256, 256, 0, stream>>>(seq, A, Bm, Cm, Dl, Spre, out);
}